// MultiProtoAsConv_8993661517889
// MI455X (gfx1250) — compile-verified
//
#include <hip/hip_runtime.h>
#include <math.h>

// ---------------------------------------------------------------------------
// MultiProtoAsConv for MI455X (gfx1250, wave32, WMMA)
// B=16, C=512, H=W=128 ; protos P=257 padded to 272 (17 tiles of 16)
// Memory-bound: ~1.08 GB mandatory traffic -> ~46us @ 23.3 TB/s.
// GEMM uses exact fp32 WMMA (V_WMMA_F32_16X16X4_F32); compute << memory floor.
// Round 2: protos staged in K-pair-interleaved layout so every WMMA A-fragment
// is one global_load_b64 (17 loads/K-step instead of 34 b32 loads).
// ---------------------------------------------------------------------------

typedef __attribute__((ext_vector_type(2))) float v2f;
typedef __attribute__((ext_vector_type(8))) float v8f;

#define B_    16
#define C_    512
#define H_    128
#define W_    128
#define HW_   (H_ * W_)
#define PLOC  256
#define PPAD  272          // 257 rounded up to 17*16
#define MT    17           // number of 16-row proto tiles
#define KQ_   (C_ / 4)     // 128 K-quads

// ---------------------------------------------------------------------------
// K1: 8x8 avg-pool of sup_fts into pool[b][c][cell] (cell = 0..255),
//     plus per-(b,c) masked sum for the global prototype numerator.
// grid = (C_, B_), block = 256 threads (one per pooled cell)
// ---------------------------------------------------------------------------
__global__ void __launch_bounds__(256) pool_feats_kernel(
    const float* __restrict__ sup, const float* __restrict__ mask,
    float* __restrict__ pool, float* __restrict__ gnum) {
  const int c = blockIdx.x, b = blockIdx.y;
  const int t = threadIdx.x;
  const int ty = t >> 4, tx = t & 15;
  const float* sp = sup + ((size_t)(b * C_ + c)) * HW_;
  const float* mp = mask + (size_t)b * HW_;
  const int base = (ty * 8) * W_ + tx * 8;

  float s = 0.0f, ms = 0.0f;
#pragma unroll
  for (int r = 0; r < 8; ++r) {
#pragma unroll
    for (int q = 0; q < 8; ++q) {
      float v  = __builtin_nontemporal_load(&sp[base + r * W_ + q]);  // streamed once
      float mk = mp[base + r * W_ + q];                               // cached (reused 512x)
      s += v;
      ms = fmaf(v, mk, ms);
    }
  }
  pool[((size_t)(b * C_ + c)) * PLOC + t] = s * (1.0f / 64.0f);

  __shared__ float red[256];
  red[t] = ms;
  __syncthreads();
#pragma unroll
  for (int off = 128; off > 0; off >>= 1) {
    if (t < off) red[t] += red[t + off];
    __syncthreads();
  }
  if (t == 0) gnum[b * C_ + c] = red[0];
}

// ---------------------------------------------------------------------------
// K2: pool sup_mask / bg_mask -> validity flags (0/1) per local proto,
//     plus den[b] = sum(sup_mask). grid = (B_), block = 256
// ---------------------------------------------------------------------------
__global__ void __launch_bounds__(256) pool_masks_kernel(
    const float* __restrict__ mask, const float* __restrict__ bg,
    float* __restrict__ vflag, float* __restrict__ den) {
  const int b = blockIdx.x, t = threadIdx.x;
  const int ty = t >> 4, tx = t & 15;
  const float* mp = mask + (size_t)b * HW_;
  const float* bp = bg + (size_t)b * HW_;
  const int base = (ty * 8) * W_ + tx * 8;

  float msum = 0.0f, bsum = 0.0f;
#pragma unroll
  for (int r = 0; r < 8; ++r) {
#pragma unroll
    for (int q = 0; q < 8; ++q) {
      msum += mp[base + r * W_ + q];
      bsum += bp[base + r * W_ + q];
    }
  }
  const float pm = msum * (1.0f / 64.0f);
  const float pb = bsum * (1.0f / 64.0f);
  vflag[b * PPAD + t] = (pm > 0.5f && pb < 0.5f) ? 1.0f : 0.0f;
  if (t == 0) vflag[b * PPAD + PLOC] = 1.0f;        // global proto always valid
  if (t < PPAD - (PLOC + 1)) vflag[b * PPAD + PLOC + 1 + t] = 0.0f;  // padding rows

  __shared__ float red[256];
  red[t] = msum;
  __syncthreads();
#pragma unroll
  for (int off = 128; off > 0; off >>= 1) {
    if (t < off) red[t] += red[t + off];
    __syncthreads();
  }
  if (t == 0) den[b] = red[0];
}

// ---------------------------------------------------------------------------
// K3: finalize protos: inject global proto at p=256, zero the pad rows,
//     L2-normalize each proto column over C, and write the result in the
//     K-pair-interleaved WMMA operand layout:
//       protosI[((b*KQ_ + kq)*2 + hh)*PPAD + p] = (A[4kq+2hh][p], A[4kq+2hh+1][p])
//     so a WMMA A-fragment is one aligned b64 load per lane.
// grid = (PPAD, B_), block = 128
// ---------------------------------------------------------------------------
__global__ void __launch_bounds__(128) finalize_protos_kernel(
    const float* __restrict__ pool, const float* __restrict__ gnum,
    const float* __restrict__ den, float* __restrict__ protosI) {
  const int p = blockIdx.x, b = blockIdx.y, t = threadIdx.x;
  const float dinv = 1.0f / (den[b] + 1e-5f);

  float vals[4];
  float ss = 0.0f;
#pragma unroll
  for (int u = 0; u < 4; ++u) {
    const int c = t + u * 128;
    float v;
    if (p < PLOC)       v = pool[((size_t)(b * C_ + c)) * PLOC + p];
    else if (p == PLOC) v = gnum[b * C_ + c] * dinv;
    else                v = 0.0f;
    vals[u] = v;
    ss = fmaf(v, v, ss);
  }
  __shared__ float red[128];
  red[t] = ss;
  __syncthreads();
#pragma unroll
  for (int off = 64; off > 0; off >>= 1) {
    if (t < off) red[t] += red[t + off];
    __syncthreads();
  }
  const float invn = 1.0f / fmaxf(sqrtf(red[0]), 1e-4f);   // safe_norm EPS
#pragma unroll
  for (int u = 0; u < 4; ++u) {
    const int c = t + u * 128;
    const int kq = c >> 2;          // K-quad
    const int hh = (c >> 1) & 1;    // which pair within the quad
    const int lo = c & 1;           // element within the pair
    protosI[((((size_t)(b * KQ_ + kq)) * 2 + hh) * PPAD + p) * 2 + lo] =
        vals[u] * invn;
  }
}

// ---------------------------------------------------------------------------
// K4: main fused kernel. One wave32 per 16-pixel tile.
//   dists(272 x 16) = protos_n(272 x 512) x qry(512 x 16), fp32 WMMA K=4.
//   A-fragments: one b64 load per tile per K-step (pair-interleaved layout).
//   Query norm fused into the K-loop (sum of squares of B-fragments), then
//   accumulators scaled by 1/max(||q||,1e-4), masked softmax over 272 protos.
// grid = (HW/128, B_), block = 256 (8 waves, 8 pixel tiles)
// ---------------------------------------------------------------------------
__global__ void __launch_bounds__(256) proto_attn_kernel(
    const float* __restrict__ qry, const float* __restrict__ protosI,
    const float* __restrict__ vflag, float* __restrict__ out) {
  const int b = blockIdx.y;
  const int wave = threadIdx.x >> 5;
  const int lane = threadIdx.x & 31;
  const int j = lane & 15;      // pixel column within tile (N index)
  const int h = lane >> 4;      // half-wave selector (K/M interleave)
  const int pix0 = (blockIdx.x * 8 + wave) * 16;

  __shared__ float vfs[PPAD];
  if (threadIdx.x < PPAD) vfs[threadIdx.x] = vflag[b * PPAD + threadIdx.x];
  __syncthreads();

  // B-fragment (4x16, K x N), f32 layout:
  //   v0: lanes 0-15 K=k0   N=lane ; lanes 16-31 K=k0+2 N=lane-16
  //   v1: lanes 0-15 K=k0+1 N=lane ; lanes 16-31 K=k0+3 N=lane-16
  const float* qp = qry + ((size_t)(b * C_) + 2 * h) * HW_ + pix0 + j;
  // A-fragment (16x4, M x K): lane needs the (K=k0+2h, K=k0+2h+1) pair for
  // row p = j + m*16 -> exactly one v2f element in the interleaved layout.
  const v2f* ap = (const v2f*)protosI + ((size_t)(b * KQ_) * 2 + h) * PPAD + j;

  v8f acc[MT];
#pragma unroll
  for (int m = 0; m < MT; ++m) {
#pragma unroll
    for (int i = 0; i < 8; ++i) acc[m][i] = 0.0f;
  }

  float ssq = 0.0f;
  for (int k0 = 0; k0 < C_; k0 += 4) {
    const float b0 = __builtin_nontemporal_load(qp);        // qry streamed once
    const float b1 = __builtin_nontemporal_load(qp + HW_);
    qp += 4 * (size_t)HW_;
    ssq = fmaf(b0, b0, ssq);
    ssq = fmaf(b1, b1, ssq);
    v2f bf;
    bf[0] = b0;
    bf[1] = b1;
#pragma unroll
    for (int m = 0; m < MT; ++m) {
      const v2f af = ap[m * 16];   // one global_load_b64
      acc[m] = __builtin_amdgcn_wmma_f32_16x16x4_f32(
          /*neg_a=*/false, af, /*neg_b=*/false, bf,
          /*c_mod=*/(short)0, acc[m], /*reuse_a=*/false, /*reuse_b=*/false);
    }
    ap += 2 * (size_t)PPAD;        // advance one K-quad (2 pair-rows)
  }

  // Fused query norm: lane pair (j, j+16) holds the two halves of sum(q^2).
  ssq += __shfl_xor(ssq, 16, 32);
  const float invn = 1.0f / fmaxf(sqrtf(ssq), 1e-4f);

  // D layout: acc[m][i] for lane = (row p = m*16 + i + 8*h, col pixel = pix0+j)
  float mx = -1e30f;
#pragma unroll
  for (int m = 0; m < MT; ++m) {
#pragma unroll
    for (int i = 0; i < 8; ++i) {
      const float d = acc[m][i] * invn;
      acc[m][i] = d;
      const float vf = vfs[m * 16 + i + h * 8];
      mx = fmaxf(mx, d + (vf - 1.0f) * 1e30f);   // invalid -> ~-1e30
    }
  }
  mx = fmaxf(mx, __shfl_xor(mx, 16, 32));

  float S = 0.0f, T = 0.0f;
#pragma unroll
  for (int m = 0; m < MT; ++m) {
#pragma unroll
    for (int i = 0; i < 8; ++i) {
      const float d = acc[m][i];
      const float vf = vfs[m * 16 + i + h * 8];
      const float e = vf * __expf(d - mx);       // invalid -> exactly 0
      S += e;
      T = fmaf(e, d, T);
    }
  }
  S += __shfl_xor(S, 16, 32);
  T += __shfl_xor(T, 16, 32);

  if (h == 0) out[(size_t)b * HW_ + pix0 + j] = T / S;
}

// ---------------------------------------------------------------------------
// Launch
// inputs: 0=qry_fts (B,C,H,W) f32, 1=sup_fts, 2=sup_mask (B,1,H,W),
//         3=true_bg_mask. out: (B,1,H,W) f32.
// workspace (floats):
//   protosI : B*C*PPAD      = 2,228,224   (interleaved WMMA A operand)
//   pool    : B*C*PLOC      = 2,097,152   (flat pooled feats, K1 -> K3)
//   gnum    : B*C           = 8,192
//   den     : B             = 16
//   vflag   : B*PPAD        = 4,352
//   total  ~= 17.4 MB
// ---------------------------------------------------------------------------
extern "C" void kernel_launch(void* const* d_in, const int* in_sizes, int n_in,
                              void* d_out, int out_size, void* d_ws, size_t ws_size,
                              hipStream_t stream) {
  const float* qry   = (const float*)d_in[0];
  const float* sup   = (const float*)d_in[1];
  const float* smask = (const float*)d_in[2];
  const float* bg    = (const float*)d_in[3];
  float* out = (float*)d_out;

  float* protosI = (float*)d_ws;                           // B*C*PPAD
  float* pool    = protosI + (size_t)B_ * C_ * PPAD;       // B*C*PLOC
  float* gnum    = pool + (size_t)B_ * C_ * PLOC;          // B*C
  float* den     = gnum + (size_t)B_ * C_;                 // B
  float* vfl     = den + B_;                               // B*PPAD

  pool_feats_kernel<<<dim3(C_, B_), 256, 0, stream>>>(sup, smask, pool, gnum);
  pool_masks_kernel<<<dim3(B_), 256, 0, stream>>>(smask, bg, vfl, den);
  finalize_protos_kernel<<<dim3(PPAD, B_), 128, 0, stream>>>(pool, gnum, den, protosI);
  proto_attn_kernel<<<dim3(HW_ / 128, B_), 256, 0, stream>>>(qry, protosI, vfl, out);
}